// TransLOB_38414187495908
// MI455X (gfx1250) — compile-verified
//
#include <hip/hip_runtime.h>
#include <math.h>

#define SEQ_LEN 1024
#define CNN_OUT 14
#define DIM 15
#define HEADS 3
#define DH 5
#define BATCH 32
#define TIN 993

typedef __attribute__((ext_vector_type(2))) float v2f;
typedef __attribute__((ext_vector_type(8))) float v8f;

static __device__ __forceinline__ v8f wmma_f32(v2f a, v2f b, v8f c) {
  // D(16x16,f32) = A(16x4,f32) * B(4x16,f32) + C
  return __builtin_amdgcn_wmma_f32_16x16x4_f32(false, a, false, b, (short)0, c,
                                               false, false);
}

#define CDIV(a, b) (((a) + (b) - 1) / (b))

// -------------------- dilated conv (kernel width 2) + bias + relu ------------
__global__ void conv_relu_kernel(const float* __restrict__ in,
                                 const float* __restrict__ w,
                                 const float* __restrict__ bias,
                                 float* __restrict__ out,
                                 int Lin, int Cin, int Lout, int dil) {
  int idx = blockIdx.x * blockDim.x + threadIdx.x;
  int total = BATCH * Lout * CNN_OUT;
  if (idx >= total) return;
  int oc = idx % CNN_OUT;
  int t  = (idx / CNN_OUT) % Lout;
  int b  = idx / (CNN_OUT * Lout);
  int t0 = t - dil;
  const float* x0 = in + ((size_t)b * Lin + t0) * Cin;
  const float* x1 = in + ((size_t)b * Lin + t) * Cin;
  const float* wr = w + (size_t)oc * Cin * 2;
  bool v0 = (t0 >= 0) && (t0 < Lin);
  bool v1 = (t < Lin);
  float acc = bias[oc];
  for (int i = 0; i < Cin; ++i) {
    float xa = v0 ? x0[i] : 0.f;
    float xb = v1 ? x1[i] : 0.f;
    acc += wr[2 * i] * xa + wr[2 * i + 1] * xb;
  }
  out[idx] = fmaxf(acc, 0.f);
}

// ----------- LayerNorm over CNN_OUT channels + concat temporal enc -----------
__global__ void ln_concat_kernel(const float* __restrict__ h,
                                 const float* __restrict__ w,
                                 const float* __restrict__ b,
                                 const float* __restrict__ temp,
                                 float* __restrict__ z) {
  int idx = blockIdx.x * blockDim.x + threadIdx.x;
  if (idx >= BATCH * SEQ_LEN) return;
  int t = idx % SEQ_LEN;
  const float* hr = h + (size_t)idx * CNN_OUT;
  float m = 0.f;
  for (int c = 0; c < CNN_OUT; ++c) m += hr[c];
  m *= (1.f / CNN_OUT);
  float v = 0.f;
  for (int c = 0; c < CNN_OUT; ++c) { float d = hr[c] - m; v += d * d; }
  v *= (1.f / CNN_OUT);
  float inv = rsqrtf(v + 1e-5f);
  float* zr = z + (size_t)idx * DIM;
  for (int c = 0; c < CNN_OUT; ++c) zr[c] = (hr[c] - m) * inv * w[c] + b[c];
  zr[CNN_OUT] = temp[t];
}

// -------------------------- generic LayerNorm --------------------------------
__global__ void ln_kernel(const float* __restrict__ x, const float* __restrict__ w,
                          const float* __restrict__ b, float* __restrict__ y,
                          int rows, int dim) {
  int idx = blockIdx.x * blockDim.x + threadIdx.x;
  if (idx >= rows) return;
  const float* xr = x + (size_t)idx * dim;
  float m = 0.f;
  for (int c = 0; c < dim; ++c) m += xr[c];
  m /= dim;
  float v = 0.f;
  for (int c = 0; c < dim; ++c) { float d = xr[c] - m; v += d * d; }
  v /= dim;
  float inv = rsqrtf(v + 1e-5f);
  float* yr = y + (size_t)idx * dim;
  for (int c = 0; c < dim; ++c) yr[c] = (xr[c] - m) * inv * w[c] + b[c];
}

// ------------- generic row-major linear: y = act(x W^T + b) (+res) -----------
// act: 0=none, 1=gelu(exact), 2=relu
__global__ void linear_kernel(const float* __restrict__ x,
                              const float* __restrict__ W,
                              const float* __restrict__ bias,
                              const float* __restrict__ residual,
                              float* __restrict__ y,
                              int rows, int in_dim, int out_dim, int act) {
  int idx = blockIdx.x * blockDim.x + threadIdx.x;
  if (idx >= rows * out_dim) return;
  int r = idx / out_dim;
  int j = idx % out_dim;
  const float* xr = x + (size_t)r * in_dim;
  const float* wj = W + (size_t)j * in_dim;
  float acc = bias[j];
  for (int k = 0; k < in_dim; ++k) acc += xr[k] * wj[k];
  if (act == 1) acc = 0.5f * acc * (1.f + erff(acc * 0.7071067811865476f));
  else if (act == 2) acc = fmaxf(acc, 0.f);
  if (residual) acc += residual[idx];
  y[idx] = acc;
}

// ------------------ flash attention, fp32 WMMA 16x16x4 -----------------------
// qkv layout: [b, n, s(3), h(3), d(5)] ; out: [b, n, h*5+d]
// One wave per (b, h, 16-row query tile). wave32.
__global__ __launch_bounds__(128)
void attn_kernel(const float* __restrict__ qkv, float* __restrict__ out) {
  __shared__ float ldsP[4][256];  // per-wave 16x16 P staging
  const int lane = threadIdx.x & 31;
  const int wib  = threadIdx.x >> 5;
  const int waveId = blockIdx.x * 4 + wib;
  const int tileI = waveId & 63;            // query row tile
  const int h = (waveId >> 6) % HEADS;
  const int b = waveId / (64 * HEADS);

  float* pbuf = ldsP[wib];
  const int half = lane >> 4;               // lane group (0: 0-15, 1: 16-31)
  const int l16  = lane & 15;
  const int f0   = 2 * half;                // feat pair {0,1} or {2,3}
  const float scale = 0.44721359549995794f; // 1/sqrt(5)
  const int dclamp = (l16 < DH) ? l16 : (DH - 1);  // always-valid dim index
  const bool dvalid = (l16 < DH);

  const float* base = qkv + (size_t)b * SEQ_LEN * (3 * DIM);

  // Q tile in A-layout: lane l16 = row, VGPR r = K(feat) 2*half+r per 4-chunk
  const int qrow = tileI * 16 + l16;
  const float* qp = base + (size_t)qrow * (3 * DIM) + h * DH;  // s=0
  v2f a0, a1;
  {
    a0[0] = qp[f0];
    a0[1] = qp[f0 + 1];
    float q4 = qp[4];                  // unconditional: always in-bounds
    a1[0] = (half == 0) ? q4 : 0.f;    // feats {4,5}/{6,7}; only 4 valid
    a1[1] = 0.f;
  }

  v8f o = {0.f, 0.f, 0.f, 0.f, 0.f, 0.f, 0.f, 0.f};
  float m[8], l[8];
#pragma unroll
  for (int v = 0; v < 8; ++v) { m[v] = -3.0e38f; l[v] = 0.f; }

  auto process_tile = [&](int j, bool masked) {
    // K tile as B (=K^T, feat x key): VGPR r / half h = feat row 2h+r, col=key
    const float* kp = base + (size_t)(j * 16 + l16) * (3 * DIM) + DIM + h * DH;
    v2f bk0, bk1;
    {
      bk0[0] = kp[f0];
      bk0[1] = kp[f0 + 1];
      float k4 = kp[4];                // unconditional: always in-bounds
      bk1[0] = (half == 0) ? k4 : 0.f;
      bk1[1] = 0.f;
    }
    v8f s = {0.f, 0.f, 0.f, 0.f, 0.f, 0.f, 0.f, 0.f};
    s = wmma_f32(a0, bk0, s);   // K-dim 0..3
    s = wmma_f32(a1, bk1, s);   // K-dim 4..7 (5..7 zero-padded)

    // scale (+ causal mask only on the diagonal tile)
    float sv[8];
    if (masked) {
      const int col = j * 16 + l16;
#pragma unroll
      for (int v = 0; v < 8; ++v) {
        int row = tileI * 16 + v + 8 * half;
        float t = s[v] * scale;
        sv[v] = (col <= row) ? t : -3.0e38f;
      }
    } else {
#pragma unroll
      for (int v = 0; v < 8; ++v) sv[v] = s[v] * scale;
    }
    // row-wise max across the 16-lane group
    float mt[8];
#pragma unroll
    for (int v = 0; v < 8; ++v) mt[v] = sv[v];
#pragma unroll
    for (int off = 1; off < 16; off <<= 1) {
#pragma unroll
      for (int v = 0; v < 8; ++v) mt[v] = fmaxf(mt[v], __shfl_xor(mt[v], off, 32));
    }
    float fac[8];
#pragma unroll
    for (int v = 0; v < 8; ++v) {
      float nm = fmaxf(m[v], mt[v]);
      fac[v] = expf(m[v] - nm);
      m[v] = nm;
    }
    float p[8], rs[8];
#pragma unroll
    for (int v = 0; v < 8; ++v) { p[v] = expf(sv[v] - m[v]); rs[v] = p[v]; }
#pragma unroll
    for (int off = 1; off < 16; off <<= 1) {
#pragma unroll
      for (int v = 0; v < 8; ++v) rs[v] += __shfl_xor(rs[v], off, 32);
    }
#pragma unroll
    for (int v = 0; v < 8; ++v) { l[v] = l[v] * fac[v] + rs[v]; o[v] = o[v] * fac[v]; }

    // P: C-layout -> LDS -> A-layout (same-wave DS ops are in-order; no barrier)
#pragma unroll
    for (int v = 0; v < 8; ++v) pbuf[(v + 8 * half) * 16 + l16] = p[v];
#pragma unroll
    for (int kc = 0; kc < 4; ++kc) {
      const int kb = kc * 4 + 2 * half;     // key sub-index of this K-pair
      v2f pa, bv;
      pa[0] = pbuf[l16 * 16 + kb];
      pa[1] = pbuf[l16 * 16 + kb + 1];
      // V rows as B: row = key kb(+1), col = head-dim l16 (>=5 -> 0)
      // clamp dim index so the load address is always valid; select afterwards
      const float* vp = base + (size_t)(j * 16 + kb) * (3 * DIM) + 2 * DIM + h * DH;
      float v0 = vp[dclamp];
      float v1 = vp[3 * DIM + dclamp];
      bv[0] = dvalid ? v0 : 0.f;
      bv[1] = dvalid ? v1 : 0.f;
      o = wmma_f32(pa, bv, o);
    }
  };

  for (int j = 0; j < tileI; ++j) process_tile(j, false);
  process_tile(tileI, true);

  // normalize and scatter (cols 0..4 hold head-dim values)
#pragma unroll
  for (int v = 0; v < 8; ++v) {
    if (dvalid) {
      int row = tileI * 16 + v + 8 * half;
      out[((size_t)b * SEQ_LEN + row) * DIM + h * DH + l16] = o[v] / l[v];
    }
  }
}

// ----------------------------- final softmax (3) -----------------------------
__global__ void softmax3_kernel(const float* __restrict__ logi,
                                float* __restrict__ out) {
  int b = threadIdx.x;
  if (b >= BATCH) return;
  float a0 = logi[b * 3 + 0], a1 = logi[b * 3 + 1], a2 = logi[b * 3 + 2];
  float mx = fmaxf(a0, fmaxf(a1, a2));
  float e0 = expf(a0 - mx), e1 = expf(a1 - mx), e2 = expf(a2 - mx);
  float inv = 1.f / (e0 + e1 + e2);
  out[b * 3 + 0] = e0 * inv;
  out[b * 3 + 1] = e1 * inv;
  out[b * 3 + 2] = e2 * inv;
}

// ============================== host orchestration ===========================
extern "C" void kernel_launch(void* const* d_in, const int* in_sizes, int n_in,
                              void* d_out, int out_size, void* d_ws, size_t ws_size,
                              hipStream_t stream) {
  (void)in_sizes; (void)n_in; (void)out_size; (void)ws_size;
  const float* x = (const float*)d_in[0];
  const float* conv_w[5];
  const float* conv_b[5];
  for (int i = 0; i < 5; ++i) {
    conv_w[i] = (const float*)d_in[1 + i];
    conv_b[i] = (const float*)d_in[6 + i];
  }
  const float* nin_w = (const float*)d_in[11];
  const float* nin_b = (const float*)d_in[12];
  const float* temp  = (const float*)d_in[13];
  const float* cls_w1 = (const float*)d_in[34];
  const float* cls_b1 = (const float*)d_in[35];
  const float* cls_w2 = (const float*)d_in[36];
  const float* cls_b2 = (const float*)d_in[37];

  float* ws = (float*)d_ws;
  float* bufA = ws;                  // 32*1024*14
  float* bufB = bufA + 458752;       // 32*1024*14
  float* z    = bufB + 458752;       // 32*1024*15
  float* qkv  = z    + 491520;       // 32*1024*45
  float* attb = qkv  + 1474560;      // 32*1024*15
  float* lnb  = attb + 491520;       // 32*1024*15
  float* mlp  = lnb  + 491520;       // 32*1024*60
  float* clsh = mlp  + 1966080;      // 32*64
  float* logi = clsh + 2048;         // 32*3

  // dilated conv stack: lengths 993 -> 994 -> 996 -> 1000 -> 1008 -> 1024
  int Ls[6] = {TIN, 994, 996, 1000, 1008, 1024};
  const float* cin = x;
  float* couts[5] = {bufA, bufB, bufA, bufB, bufA};
  int Cin = 40;
  for (int i = 0; i < 5; ++i) {
    int d = 1 << i;
    int total = BATCH * Ls[i + 1] * CNN_OUT;
    conv_relu_kernel<<<CDIV(total, 256), 256, 0, stream>>>(
        cin, conv_w[i], conv_b[i], couts[i], Ls[i], Cin, Ls[i + 1], d);
    cin = couts[i];
    Cin = CNN_OUT;
  }

  ln_concat_kernel<<<CDIV(BATCH * SEQ_LEN, 256), 256, 0, stream>>>(
      bufA, nin_w, nin_b, temp, z);

  const int rows = BATCH * SEQ_LEN;
  for (int blk = 0; blk < 2; ++blk) {
    const float* qkv_w  = (const float*)d_in[14 + blk * 10 + 0];
    const float* qkv_b  = (const float*)d_in[14 + blk * 10 + 1];
    const float* proj_w = (const float*)d_in[14 + blk * 10 + 2];
    const float* proj_b = (const float*)d_in[14 + blk * 10 + 3];
    const float* n2_w   = (const float*)d_in[14 + blk * 10 + 4];
    const float* n2_b   = (const float*)d_in[14 + blk * 10 + 5];
    const float* fc1_w  = (const float*)d_in[14 + blk * 10 + 6];
    const float* fc1_b  = (const float*)d_in[14 + blk * 10 + 7];
    const float* fc2_w  = (const float*)d_in[14 + blk * 10 + 8];
    const float* fc2_b  = (const float*)d_in[14 + blk * 10 + 9];

    linear_kernel<<<CDIV(rows * 45, 256), 256, 0, stream>>>(
        z, qkv_w, qkv_b, nullptr, qkv, rows, DIM, 3 * DIM, 0);
    attn_kernel<<<(BATCH * HEADS * 64) / 4, 128, 0, stream>>>(qkv, attb);
    linear_kernel<<<CDIV(rows * DIM, 256), 256, 0, stream>>>(
        attb, proj_w, proj_b, z, z, rows, DIM, DIM, 0);
    ln_kernel<<<CDIV(rows, 256), 256, 0, stream>>>(z, n2_w, n2_b, lnb, rows, DIM);
    linear_kernel<<<CDIV(rows * 60, 256), 256, 0, stream>>>(
        lnb, fc1_w, fc1_b, nullptr, mlp, rows, DIM, 60, 1);
    linear_kernel<<<CDIV(rows * DIM, 256), 256, 0, stream>>>(
        mlp, fc2_w, fc2_b, z, z, rows, 60, DIM, 0);
  }

  // classifier
  linear_kernel<<<CDIV(BATCH * 64, 256), 256, 0, stream>>>(
      z, cls_w1, cls_b1, nullptr, clsh, BATCH, SEQ_LEN * DIM, 64, 2);
  linear_kernel<<<1, 128, 0, stream>>>(
      clsh, cls_w2, cls_b2, nullptr, logi, BATCH, 64, 3, 0);
  softmax3_kernel<<<1, 32, 0, stream>>>(logi, (float*)d_out);
}